// ThresholdedMSELoss_386547057265
// MI455X (gfx1250) — compile-verified
//
#include <hip/hip_runtime.h>

// ThresholdedMSELoss == plain MSE (all branches compute (x-y)^2).
// Memory-bound streaming reduction for MI455X (gfx1250, wave32):
//   pass 1: 1024 blocks x 256 thr, float4 NT loads, fma accumulate,
//           wave32 shuffle reduce + LDS cross-wave reduce -> d_ws[block]
//   pass 2: one wave folds 1024 partials to 32 lane sums, then a
//           v_wmma_f32_16x16x4_f32 (A = lane partials, B = ones) performs
//           the final cross-lane sum; lane 0 writes mean.

typedef __attribute__((ext_vector_type(4))) float v4f;
typedef __attribute__((ext_vector_type(2))) float v2f;
typedef __attribute__((ext_vector_type(8))) float v8f;

#define WAVE    32
#define BLK     256
#define NWAVES  (BLK / WAVE)
#define NBLOCKS 1024

__global__ __launch_bounds__(BLK) void mse_partial_kernel(
    const float* __restrict__ pred, const float* __restrict__ lab,
    float* __restrict__ partials, long long n) {
  const long long n4     = n >> 2;                      // # of float4 chunks
  const long long stride = (long long)gridDim.x * blockDim.x;
  const long long tid0   = (long long)blockIdx.x * blockDim.x + threadIdx.x;

  const v4f* __restrict__ p4 = (const v4f*)pred;
  const v4f* __restrict__ l4 = (const v4f*)lab;

  float s = 0.0f;
  // Main stream: 128-bit non-temporal loads (single-use data; don't pollute L2).
  for (long long i = tid0; i < n4; i += stride) {
    v4f a = __builtin_nontemporal_load(&p4[i]);
    v4f b = __builtin_nontemporal_load(&l4[i]);
    v4f d = a - b;
    s = fmaf(d.x, d.x, s);
    s = fmaf(d.y, d.y, s);
    s = fmaf(d.z, d.z, s);
    s = fmaf(d.w, d.w, s);
  }
  // Scalar tail (N = 2^25 is divisible by 4, but stay generic).
  for (long long i = (n4 << 2) + tid0; i < n; i += stride) {
    float d = pred[i] - lab[i];
    s = fmaf(d, d, s);
  }

  // wave32 reduction
  #pragma unroll
  for (int off = WAVE / 2; off > 0; off >>= 1)
    s += __shfl_xor(s, off, WAVE);

  __shared__ float lds[NWAVES];
  const int lane = threadIdx.x & (WAVE - 1);
  const int wave = threadIdx.x / WAVE;
  if (lane == 0) lds[wave] = s;
  __syncthreads();

  if (threadIdx.x == 0) {
    float bs = 0.0f;
    #pragma unroll
    for (int w = 0; w < NWAVES; ++w) bs += lds[w];
    partials[blockIdx.x] = bs;
  }
}

// One full wave (EXEC all-1s at the WMMA). Folds `nparts` partials to 32 lane
// sums, then uses v_wmma_f32_16x16x4_f32 to sum across lanes:
//   A(16x4): lanes 0-15 supply K=0 (VGPR0) / K=1 (VGPR1);
//            lanes 16-31 supply K=2 / K=3.  We put partial in VGPR0, 0 in VGPR1
//            => A[m][0] = p_m, A[m][2] = p_{m+16}, cols 1,3 = 0.
//   B(4x16) = all ones  => D[m][n] = p_m + p_{m+16}  for every n.
// Column 0 of D lives in VGPRs 0..7 of lane 0 (M=0..7) and lane 16 (M=8..15),
// so total = sum(c[0..7])@lane0 + sum(c[0..7])@lane16.
__global__ __launch_bounds__(WAVE) void mse_final_kernel(
    const float* __restrict__ partials, int nparts,
    float* __restrict__ out, float inv_n) {
  float s = 0.0f;
  for (int i = threadIdx.x; i < nparts; i += WAVE) s += partials[i];

  v2f a; a.x = s;    a.y = 0.0f;
  v2f b; b.x = 1.0f; b.y = 1.0f;
  v8f c = {};
  c = __builtin_amdgcn_wmma_f32_16x16x4_f32(
      /*neg_a=*/false, a, /*neg_b=*/false, b,
      /*c_mod=*/(short)0, c, /*reuse_a=*/false, /*reuse_b=*/false);

  float colsum = c[0] + c[1] + c[2] + c[3] + c[4] + c[5] + c[6] + c[7];
  float total  = __shfl(colsum, 0, WAVE) + __shfl(colsum, 16, WAVE);
  if (threadIdx.x == 0) out[0] = total * inv_n;
}

extern "C" void kernel_launch(void* const* d_in, const int* in_sizes, int n_in,
                              void* d_out, int out_size, void* d_ws, size_t ws_size,
                              hipStream_t stream) {
  const float* pred = (const float*)d_in[0];
  const float* lab  = (const float*)d_in[1];
  float* out        = (float*)d_out;
  float* partials   = (float*)d_ws;   // NBLOCKS floats of scratch
  const long long n = (long long)in_sizes[0];

  mse_partial_kernel<<<NBLOCKS, BLK, 0, stream>>>(pred, lab, partials, n);
  mse_final_kernel<<<1, WAVE, 0, stream>>>(partials, NBLOCKS, out,
                                           (float)(1.0 / (double)n));
}